// ForceFolder_80891414053203
// MI455X (gfx1250) — compile-verified
//
#include <hip/hip_runtime.h>

// ForceFolder: Gaussian-mixture pair forces.
//   T=8 trajs, N=500 residues, B=34 bins.
//   out[t,j,:] = sum_i  -forces[t,i,j] * 150 * (c_j - c_i)/dist
//   forces     = sum_b  -(dist-c_b)/s^2 * w[i,j,b]*exp(-0.5((dist-c_b)/s)^2)/s
//
// MI455X strategy:
//  * Weights (34 MB) dominate traffic; resident in 192 MB L2, reused 8x across
//    trajectories -> HBM floor ~34MB / 23.3TB/s ~ 1.5us; kernel is v_exp bound.
//  * Pair dots via V_WMMA_F32_16X16X4_F32 Gram tiles: one WMMA = 16x16 block of
//    c_i . c_j; D layout hands each lane its j-column and 8 i-rows.
//  * Coordinates staged in LDS (zero-padded to 512 rows); branchless body.
//  * Register-pressure control: r-loop unrolled only 2x (two 136-B weight rows
//    in flight max) with a uniform-condition cndmask tree to pick dot[r] --
//    avoids the full-unroll spill storm seen in round 2.
//  * Deterministic: one block owns each (t, j-tile); fixed-order LDS reduction.

#define T_TRAJ 8
#define NRES   500
#define NPAD   512
#define NBINS  34
#define K_CB   150.0f
#define FCUT   40.0f

typedef __attribute__((ext_vector_type(2))) float v2f;
typedef __attribute__((ext_vector_type(8))) float v8f;

// Select d[r] for wave-uniform r: 7 v_cndmask with SGPR conditions, no scratch.
__device__ __forceinline__ float sel8(const v8f d, int r) {
    const float lo = (r & 2) ? ((r & 1) ? d[3] : d[2])
                             : ((r & 1) ? d[1] : d[0]);
    const float hi = (r & 2) ? ((r & 1) ? d[7] : d[6])
                             : ((r & 1) ? d[5] : d[4]);
    return (r & 4) ? hi : lo;
}

__global__ __launch_bounds__(512) void ForceFolder_kernel(
    const float* __restrict__ coords,   // [T, N, 3]
    const float* __restrict__ gw,       // [N(i), N(j), NBINS]
    const float* __restrict__ centres,  // [NBINS]
    const float* __restrict__ sigmas,   // [NBINS]
    float* __restrict__ out)            // [T, N, 3]
{
    __shared__ float s_cent[NBINS];
    __shared__ float s_nh[NBINS];          // -0.5 / sigma^2
    __shared__ float s_is3[NBINS];         //  1  / sigma^3
    __shared__ float s_crd[NPAD * 3];      // coords[t], zero-padded
    __shared__ float s_red[16][32][3];

    const int tid = threadIdx.x;
    const int jt  = blockIdx.x;            // 0..31 : 16-wide j tile
    const int t   = blockIdx.y;            // 0..7  : trajectory

    if (tid < NBINS) {
        const float s  = sigmas[tid];
        const float is = 1.0f / s;
        s_cent[tid] = centres[tid];
        s_nh[tid]   = -0.5f * is * is;
        s_is3[tid]  = is * is * is;
    }
    for (int idx = tid; idx < NPAD * 3; idx += 512) {
        s_crd[idx] = (idx < NRES * 3) ? coords[(size_t)t * (NRES * 3) + idx]
                                      : 0.0f;
    }
    __syncthreads();

    const int wave = tid >> 5;             // 0..15
    const int lane = tid & 31;
    const int hi   = lane >> 4;            // half-wave: 0 or 1
    const int jj   = lane & 15;
    const int j    = jt * 16 + jj;                     // <= 511, LDS-safe
    const int jw   = (j < NRES) ? j : (NRES - 1);      // clamp for weight base
    const bool jok = (j < NRES);

    // This lane's j-column coordinates (zero for padded rows -> masked later).
    const float cjx = s_crd[j * 3 + 0];
    const float cjy = s_crd[j * 3 + 1];
    const float cjz = s_crd[j * 3 + 2];
    const float nj  = cjx * cjx + cjy * cjy + cjz * cjz;

    // B fragment (4x16, K = xyz0): VGPR0 = K0 row (lanes 0-15) / K2 row
    // (lanes 16-31); VGPR1 = K1 / K3. Loop-invariant per wave.
    v2f bfrag;
    bfrag.x = hi ? cjz : cjx;
    bfrag.y = hi ? 0.0f : cjy;

    float accx = 0.0f, accy = 0.0f, accz = 0.0f;

    // 32 i-tiles of 16 cover i in [0,512); 16 waves stride them (2 each).
    #pragma unroll 1
    for (int it = wave; it < 32; it += 16) {
        // A fragment (16x4): lane L holds row M = L&15; lanes<16 carry K0,K1,
        // lanes>=16 carry K2,K3.
        const int ia = it * 16 + jj;                   // <= 511, LDS-safe
        v2f afrag;
        {
            const float ax = s_crd[ia * 3 + 0];
            const float ay = s_crd[ia * 3 + 1];
            const float az = s_crd[ia * 3 + 2];
            afrag.x = hi ? az : ax;
            afrag.y = hi ? 0.0f : ay;
        }

        v8f czero = {};
        // dot[r] = c_{i = it*16 + r + 8*hi} . c_{j = this lane's column}
        const v8f dot = __builtin_amdgcn_wmma_f32_16x16x4_f32(
            /*neg_a=*/false, afrag, /*neg_b=*/false, bfrag,
            /*c_mod=*/(short)0, czero, /*reuse_a=*/false, /*reuse_b=*/false);

        const int ibase = it * 16 + 8 * hi;

        // Partial unroll: at most two 34-float weight rows live at once.
        #pragma unroll 2
        for (int r = 0; r < 8; ++r) {
            const int i = ibase + r;                   // <= 511, LDS-safe
            // Uniform across each half-wave -> LDS broadcast.
            const float cix = s_crd[i * 3 + 0];
            const float ciy = s_crd[i * 3 + 1];
            const float ciz = s_crd[i * 3 + 2];
            const float ni  = cix * cix + ciy * ciy + ciz * ciz;

            float d2 = ni + nj - 2.0f * sel8(dot, r);
            d2 = fmaxf(d2, 0.0f);                      // kill cancellation
            float dist = __builtin_amdgcn_sqrtf(d2);
            dist = fminf(fmaxf(dist, 0.1f), FCUT);

            // 34-bin mixture; weight rows are 136 B contiguous, base element
            // index (i*500+j)*34 is even -> 8 B aligned: float2 (compiler
            // widens to b128 runs).
            const int iw = (i < NRES) ? i : (NRES - 1);
            const float2* wp =
                (const float2*)(gw + ((size_t)iw * NRES + jw) * NBINS);
            float F = 0.0f;                            // == reference 'forces'
            #pragma unroll
            for (int bb = 0; bb < NBINS / 2; ++bb) {
                const float2 w2 = wp[bb];
                const int b0 = 2 * bb, b1 = 2 * bb + 1;
                const float dfm0 = dist - s_cent[b0];
                const float g0   = __expf(dfm0 * dfm0 * s_nh[b0]);
                F -= w2.x * g0 * dfm0 * s_is3[b0];
                const float dfm1 = dist - s_cent[b1];
                const float g1   = __expf(dfm1 * dfm1 * s_nh[b1]);
                F -= w2.y * g1 * dfm1 * s_is3[b1];
            }

            // pair = -F * K * (c_j - c_i) / dist, masked branchlessly.
            const float vmask = (jok && i < NRES) ? 1.0f : 0.0f;
            const float scale =
                (-K_CB) * F * __builtin_amdgcn_rcpf(dist) * vmask;
            accx += scale * (cjx - cix);
            accy += scale * (cjy - ciy);
            accz += scale * (cjz - ciz);
        }
    }

    // Deterministic fixed-order reduction over the 32 (wave, half) slots per j.
    const int slot = wave * 2 + hi;
    s_red[jj][slot][0] = accx;
    s_red[jj][slot][1] = accy;
    s_red[jj][slot][2] = accz;
    __syncthreads();

    if (tid < 16 * 3) {
        const int jl = tid / 3;
        const int c  = tid % 3;
        float s = 0.0f;
        #pragma unroll
        for (int k = 0; k < 32; ++k) s += s_red[jl][k][c];
        const int jg = jt * 16 + jl;
        if (jg < NRES) out[((size_t)t * NRES + jg) * 3 + c] = s;
    }
}

extern "C" void kernel_launch(void* const* d_in, const int* in_sizes, int n_in,
                              void* d_out, int out_size, void* d_ws, size_t ws_size,
                              hipStream_t stream) {
    (void)in_sizes; (void)n_in; (void)out_size; (void)d_ws; (void)ws_size;
    const float* coords  = (const float*)d_in[0];  // [8,500,3]
    const float* gw      = (const float*)d_in[1];  // [500,500,34]
    const float* centres = (const float*)d_in[2];  // [34]
    const float* sigmas  = (const float*)d_in[3];  // [34]
    float* out = (float*)d_out;                    // [8,500,3]

    dim3 grid(32 /*j tiles*/, T_TRAJ /*t*/);
    ForceFolder_kernel<<<grid, 512, 0, stream>>>(coords, gw, centres, sigmas, out);
}